// MultiHeadedCodebook_9113920602162
// MI455X (gfx1250) — compile-verified
//
#include <hip/hip_runtime.h>

typedef __attribute__((ext_vector_type(2))) float v2f;
typedef __attribute__((ext_vector_type(8))) float v8f;

#define B_   8
#define L_   4096
#define DIM_ 1024
#define H_   8
#define DH_  128
#define M_   512
#define R_        (B_ * L_)        // 32768 rows
#define ZQ_ELEMS  ((size_t)R_ * DIM_)
#define IDX_ELEMS ((size_t)R_ * H_)
#define LDS_STRIDE 132             // 128 + 4 pad: rows hit distinct LDS banks

// ---------------------------------------------------------------------------
// Kernel 1: per-code squared norms  csq[h][m] = sum_d cb[h][m][d]^2
// ---------------------------------------------------------------------------
__global__ void vq_csq_kernel(const float* __restrict__ cb,
                              float* __restrict__ csq) {
    int i = blockIdx.x * blockDim.x + threadIdx.x;   // 0 .. H*M-1
    if (i >= H_ * M_) return;
    const float* r = cb + (size_t)i * DH_;
    float s = 0.0f;
#pragma unroll 4
    for (int d = 0; d < DH_; d += 4) {
        float4 v = *(const float4*)(r + d);
        s += v.x * v.x + v.y * v.y + v.z * v.z + v.w * v.w;
    }
    csq[i] = s;
}

// ---------------------------------------------------------------------------
// Kernel 2: main VQ kernel.
// Grid: (R/16, H). Block: 128 threads = 4 waves (wave32).
// Each workgroup: 16 z-rows x 1 head x all 512 codes.
// Wave w owns codes [w*128, w*128+128) = 8 tiles of 16 codes.
// Cross term per code tile: 32 chained v_wmma_f32_16x16x4_f32 (K=128).
// ---------------------------------------------------------------------------
__launch_bounds__(128)
__global__ void vq_main_kernel(const float* __restrict__ z,
                               const float* __restrict__ cb,
                               const float* __restrict__ csq,
                               float* __restrict__ zq_out,
                               float* __restrict__ idx_out,
                               float* __restrict__ dist_out) {
    __shared__ __align__(16) float s_z[16 * LDS_STRIDE];
    __shared__ float s_zsq[16];
    __shared__ float s_min[4][16];
    __shared__ int   s_arg[4][16];
    __shared__ int   s_idx[16];

    const int tid      = threadIdx.x;
    const int lane     = tid & 31;
    const int wave     = tid >> 5;
    const int row_base = blockIdx.x << 4;   // 16 rows per tile
    const int h        = blockIdx.y;

    if (tid < 16) s_zsq[tid] = 0.0f;
    __syncthreads();

    // ---- cooperative, coalesced load of z tile into LDS + row sumsq -------
    // 16 rows x 128 floats = 512 float4 chunks; thread t takes chunks t+128*j.
    const float* zbase = z + (size_t)row_base * DIM_ + (size_t)h * DH_;
#pragma unroll
    for (int j = 0; j < 4; ++j) {
        int c  = tid + (j << 7);     // 0..511
        int r  = c >> 5;             // row in tile
        int c4 = c & 31;             // float4 column
        float4 v = *(const float4*)(zbase + (size_t)r * DIM_ + c4 * 4);
        *(float4*)&s_z[r * LDS_STRIDE + c4 * 4] = v;
        atomicAdd(&s_zsq[r], v.x * v.x + v.y * v.y + v.z * v.z + v.w * v.w);
    }
    __syncthreads();

    // ---- preload all 32 B-fragments (z^T, K x 16 rows) from LDS -----------
    // B layout (f32 4x16): lanes 0-15 hold column n=lane with K=0,1 in the
    // 2 VGPRs; lanes 16-31 hold the same columns with K=2,3.
    const int l_lo   = lane & 15;
    const int k_half = (lane >> 4) << 1;   // 0 or 2
    v2f bfrag[32];
#pragma unroll
    for (int kk = 0; kk < 32; ++kk)
        bfrag[kk] = *(const v2f*)&s_z[l_lo * LDS_STRIDE + (kk << 2) + k_half];

    const float zsq_l = s_zsq[l_lo];
    const float* cb_h  = cb  + (size_t)h * M_ * DH_;
    const float* csq_h = csq + (size_t)h * M_;

    float best_d = 3.402823466e38f;
    int   best_i = 0;

    // ---- 8 code tiles per wave -------------------------------------------
    for (int ct = 0; ct < 8; ++ct) {
        const int m_base = (wave << 7) + (ct << 4);
        // A layout (f32 16x4): lanes 0-15 hold row m=m_base+lane, K=0,1;
        // lanes 16-31 same rows, K=2,3.  Each lane streams float2 per k-step.
        const float* a_ptr = cb_h + (size_t)(m_base + l_lo) * DH_ + k_half;

        v8f acc = {};
#pragma unroll
        for (int kk = 0; kk < 32; ++kk) {
            v2f a = *(const v2f*)(a_ptr + (kk << 2));
            acc = __builtin_amdgcn_wmma_f32_16x16x4_f32(
                false, a, false, bfrag[kk], (short)0, acc, false, false);
        }

        // D layout: lane (0-15) -> column l=lane, VGPR v -> code m_base+v;
        // lane (16-31) -> column l=lane-16, VGPR v -> code m_base+8+v.
        const int m0 = m_base + ((lane >> 4) << 3);
        float4 cs0 = *(const float4*)(csq_h + m0);
        float4 cs1 = *(const float4*)(csq_h + m0 + 4);
        float cs[8] = {cs0.x, cs0.y, cs0.z, cs0.w, cs1.x, cs1.y, cs1.z, cs1.w};
#pragma unroll
        for (int v = 0; v < 8; ++v) {
            float d = fmaxf(zsq_l + cs[v] - 2.0f * acc[v], 0.0f);
            if (d < best_d) { best_d = d; best_i = m0 + v; }  // ascending m
        }
    }

    // ---- combine lane pair (l, l+16): codes +0..7 vs +8..15 ---------------
    {
        float od = __shfl_xor(best_d, 16, 32);
        int   oi = __shfl_xor(best_i, 16, 32);
        if (od < best_d || (od == best_d && oi < best_i)) {
            best_d = od; best_i = oi;
        }
    }
    if (lane < 16) { s_min[wave][lane] = best_d; s_arg[wave][lane] = best_i; }
    __syncthreads();

    // ---- combine 4 waves (ascending m ranges -> first-match tie-break) ----
    if (wave == 0 && lane < 16) {
        float bd = s_min[0][lane];
        int   bi = s_arg[0][lane];
#pragma unroll
        for (int w = 1; w < 4; ++w) {
            float d  = s_min[w][lane];
            int   i2 = s_arg[w][lane];
            if (d < bd || (d == bd && i2 < bi)) { bd = d; bi = i2; }
        }
        s_idx[lane] = bi;
        size_t row = (size_t)(row_base + lane);
        idx_out [row * H_ + h] = (float)bi;   // d_out is a uniform f32 buffer
        dist_out[row * H_ + h] = bd;
    }
    __syncthreads();

    // ---- emit z_q = z + (e - z), matching straight-through fp32 rounding --
    float* zq_base = zq_out + (size_t)row_base * DIM_ + (size_t)h * DH_;
#pragma unroll
    for (int j = 0; j < 4; ++j) {
        int c  = tid + (j << 7);
        int r  = c >> 5;
        int c4 = c & 31;
        int m  = s_idx[r];
        float4 e  = *(const float4*)(cb_h + (size_t)m * DH_ + c4 * 4);
        float4 zv = *(const float4*)&s_z[r * LDS_STRIDE + c4 * 4];
        float4 o;
        o.x = zv.x + (e.x - zv.x);
        o.y = zv.y + (e.y - zv.y);
        o.z = zv.z + (e.z - zv.z);
        o.w = zv.w + (e.w - zv.w);
        *(float4*)(zq_base + (size_t)r * DIM_ + c4 * 4) = o;
    }
}

// ---------------------------------------------------------------------------
extern "C" void kernel_launch(void* const* d_in, const int* in_sizes, int n_in,
                              void* d_out, int out_size, void* d_ws, size_t ws_size,
                              hipStream_t stream) {
    (void)in_sizes; (void)n_in; (void)out_size; (void)ws_size;
    const float* z  = (const float*)d_in[0];   // [B, L, DIM] f32
    const float* cb = (const float*)d_in[1];   // [H, M, DH]  f32

    float* out      = (float*)d_out;
    float* zq       = out;                       // [B, L, DIM]
    float* idx_out  = out + ZQ_ELEMS;            // [B, L, H] (as float)
    float* dist_out = idx_out + IDX_ELEMS;       // [B, L, H]

    float* csq = (float*)d_ws;                   // H*M floats scratch

    vq_csq_kernel<<<(H_ * M_ + 255) / 256, 256, 0, stream>>>(cb, csq);

    dim3 grid(R_ / 16, H_);
    vq_main_kernel<<<grid, 128, 0, stream>>>(z, cb, csq, zq, idx_out, dist_out);
}